// MaskedModelLogitFomatter_35914516529657
// MI455X (gfx1250) — compile-verified
//
#include <hip/hip_runtime.h>

// MaskedModelLogitFormatter: out[s,p,t] = logits[s,p,t] * table[seq[s,p], t]
// Pure memory-bound row-gather + multiply. ~1.5 GiB traffic -> ~66us @ 23.3 TB/s.
// CDNA5 path used: GLOBAL_LOAD_ASYNC_TO_LDS_B128 (ASYNCcnt) to stage the gathered
// table row chunk into LDS, overlapped with non-temporal b128 streaming of logits.

typedef float __attribute__((ext_vector_type(4))) v4f;
typedef int   __attribute__((ext_vector_type(4))) v4i;

// Pointer types matching the builtin's signature (param 0 = addrspace(1) int4*,
// param 1 = addrspace(3) int4*).
typedef __attribute__((address_space(1))) v4i* g_v4i_ptr;
typedef __attribute__((address_space(3))) v4i* l_v4i_ptr;

#define BLOCK 256
#define ITERS 4
#define CHUNK (BLOCK * 4 * ITERS)   // 4096 floats = 16 KB per block

#if defined(__has_builtin)
#if __has_builtin(__builtin_amdgcn_global_load_async_to_lds_b128)
#define HAVE_ASYNC_B128 1
#endif
#endif

__device__ __forceinline__ void async_copy_b128(const float* __restrict__ g,
                                                float* __restrict__ l) {
#if defined(HAVE_ASYNC_B128)
  __builtin_amdgcn_global_load_async_to_lds_b128(
      (g_v4i_ptr)(void*)g, (l_v4i_ptr)(void*)l, /*offset=*/0, /*cpol=*/0);
#else
  unsigned lds_addr = (unsigned)(unsigned long long)(const void*)l;
  asm volatile("global_load_async_to_lds_b128 %0, %1, off"
               :
               : "v"(lds_addr), "v"(g)
               : "memory");
#endif
}

__device__ __forceinline__ void wait_async_zero() {
#if defined(__has_builtin) && __has_builtin(__builtin_amdgcn_s_wait_asynccnt)
  __builtin_amdgcn_s_wait_asynccnt(0);
#else
  asm volatile("s_wait_asynccnt 0x0" ::: "memory");
#endif
}

__global__ void __launch_bounds__(BLOCK)
masked_logit_formatter_kernel(const float* __restrict__ logits,
                              const int* __restrict__ seq,
                              const float* __restrict__ table,
                              float* __restrict__ out,
                              int toShift,     // log2(To)
                              int cprShift) {  // log2(To / CHUNK)
  __shared__ float lds_tab[CHUNK];

  // To and CHUNK are powers of two -> pure shift/mask, no SALU division.
  const unsigned bid = blockIdx.x;
  const int row   = (int)(bid >> cprShift);              // [0, S*P)
  const int chunk = (int)(bid & ((1u << cprShift) - 1u));

  const int t = seq[row];                                // block-uniform -> s_load

  const size_t rowBase = ((size_t)row << toShift) + (size_t)chunk * CHUNK;
  const size_t tabBase = ((size_t)t   << toShift) + (size_t)chunk * CHUNK;

  const float* __restrict__ lg = logits + rowBase;
  const float* __restrict__ tb = table  + tabBase;
  float* __restrict__ op = out + rowBase;

  const int tid = (int)threadIdx.x;

  // Kick off async gather of the table chunk into LDS (per-lane b128 copies).
  // Each thread copies exactly the 4 float4s it will later consume, so no
  // cross-thread barrier is needed — only s_wait_asynccnt 0.
#pragma unroll
  for (int i = 0; i < ITERS; ++i) {
    const int off = (i * BLOCK + tid) * 4;
    async_copy_b128(tb + off, lds_tab + off);
  }

  // While the async copies are in flight, stream the logits (non-temporal:
  // touched exactly once, don't pollute caches).
  v4f l[ITERS];
#pragma unroll
  for (int i = 0; i < ITERS; ++i) {
    const int off = (i * BLOCK + tid) * 4;
    l[i] = __builtin_nontemporal_load((const v4f*)(lg + off));
  }

  wait_async_zero();

  // Multiply against the staged table row and stream results out.
#pragma unroll
  for (int i = 0; i < ITERS; ++i) {
    const int off = (i * BLOCK + tid) * 4;
    v4f m = *(const v4f*)(lds_tab + off);              // ds_load_b128
    v4f r = l[i] * m;
    __builtin_nontemporal_store(r, (v4f*)(op + off));  // global_store_b128 NT
  }
}

static inline int ilog2(unsigned x) {
  int s = 0;
  while ((1u << s) < x) ++s;
  return s;
}

extern "C" void kernel_launch(void* const* d_in, const int* in_sizes, int n_in,
                              void* d_out, int out_size, void* d_ws, size_t ws_size,
                              hipStream_t stream) {
  const float* logits = (const float*)d_in[0];   // [S,P,To] f32
  const int*   seq    = (const int*)d_in[1];     // [S,P]    int
  const float* table  = (const float*)d_in[2];   // [Ti,To]  f32
  float* out = (float*)d_out;

  const int rows = in_sizes[1];                  // S*P = 8192
  const int To   = in_sizes[0] / rows;           // 16384 (power of two, mult of CHUNK)
  const int chunksPerRow = To / CHUNK;           // 4

  const int toShift  = ilog2((unsigned)To);
  const int cprShift = ilog2((unsigned)chunksPerRow);

  dim3 grid((unsigned)(rows * chunksPerRow));
  dim3 block(BLOCK);
  masked_logit_formatter_kernel<<<grid, block, 0, stream>>>(
      logits, seq, table, out, toShift, cprShift);
}